// SpatialGate_56848187130353
// MI455X (gfx1250) — compile-verified
//
#include <hip/hip_runtime.h>
#include <hip/hip_bf16.h>

typedef float v2f __attribute__((ext_vector_type(2)));
typedef float v8f __attribute__((ext_vector_type(8)));

#define B_    4
#define C_    64
#define D_    32
#define H_    64
#define W_    64
#define HW    (H_ * W_)          // 4096
#define DHW   (D_ * HW)          // 131072
#define CDHW  (C_ * DHW)         // 8388608
#define TW    24                 // padded LDS tile row stride (cols 0..21 used)

// ---------------------------------------------------------------------------
// Kernel 1: channel-wise max and mean over C=64.  Pure bandwidth, float4.
// ---------------------------------------------------------------------------
__global__ __launch_bounds__(256) void sg_reduce_maxmean(
    const float* __restrict__ x, float* __restrict__ xmax, float* __restrict__ xmean) {
  long t = (long)blockIdx.x * blockDim.x + threadIdx.x;   // 131072 threads
  long f = t * 4;                                          // pos in [0, B*DHW)
  int  b = (int)(f / DHW);
  int  q = (int)(f % DHW);
  const float* xp = x + (long)b * CDHW + q;

  float4 v = *(const float4*)xp;
  float mx0 = v.x, mx1 = v.y, mx2 = v.z, mx3 = v.w;
  float s0 = v.x, s1 = v.y, s2 = v.z, s3 = v.w;
  for (int c = 1; c < C_; ++c) {
    float4 u = *(const float4*)(xp + (long)c * DHW);
    mx0 = fmaxf(mx0, u.x); mx1 = fmaxf(mx1, u.y);
    mx2 = fmaxf(mx2, u.z); mx3 = fmaxf(mx3, u.w);
    s0 += u.x; s1 += u.y; s2 += u.z; s3 += u.w;
  }
  const float inv = 1.0f / (float)C_;
  float4 mo; mo.x = mx0; mo.y = mx1; mo.z = mx2; mo.w = mx3;
  float4 me; me.x = s0 * inv; me.y = s1 * inv; me.z = s2 * inv; me.w = s3 * inv;
  *(float4*)(xmax  + (long)b * DHW + q) = mo;
  *(float4*)(xmean + (long)b * DHW + q) = me;
}

// ---------------------------------------------------------------------------
// Kernel 2: 7x7x7 conv (2->1) + bias + sigmoid via Toeplitz WMMA along H.
// One wave (32 threads) per 16x16 (h,w) output tile of one (b,d) slice.
// Out[16h x 16w] += T_c,kd,kw[16x24] x In[24x16], K split into 6 chunks of 4.
// A layout (16x4 f32): lane m=lane&15, K slots {kgrp,kgrp+1}, kgrp=(lane>=16)?2:0
// B layout (4x16 f32): same per-lane K slots, n=lane&15
// D layout (16x16 f32): v[r] holds M=r+(lane>=16?8:0), N=lane&15
// ---------------------------------------------------------------------------
__global__ __launch_bounds__(32) void sg_conv_sigmoid(
    const float* __restrict__ xmax, const float* __restrict__ xmean,
    const float* __restrict__ wgt, const float* __restrict__ bias,
    float* __restrict__ scale) {
  __shared__ float lw[2 * 7 * 7 * 7];   // 686 conv weights
  __shared__ float tile[TW * TW];       // 24 rows x 24 stride; rows 22,23 forced 0

  const int lane = threadIdx.x;
  const int bid  = blockIdx.x;                   // 2048 tiles
  const int wt = bid & 3, ht = (bid >> 2) & 3;
  const int d  = (bid >> 4) & 31, b = bid >> 9;
  const int hbase = ht * 16, wbase = wt * 16;

  for (int i = lane; i < 686; i += 32) lw[i] = wgt[i];
  __syncthreads();

  const int m    = lane & 15;                    // A row / D N-index
  const int n    = m;                            // B column
  const int kgrp = (lane >> 4) << 1;             // 0 (lanes 0-15) or 2 (16-31)

  v8f acc = {0.f, 0.f, 0.f, 0.f, 0.f, 0.f, 0.f, 0.f};

  for (int c = 0; c < 2; ++c) {
    const float* plane_b = (c == 0 ? xmax : xmean) + (long)b * DHW;
    for (int kd = 0; kd < 7; ++kd) {
      const int dz = d + kd - 3;
      if (dz < 0 || dz >= D_) continue;          // uniform across wave
      const float* plane = plane_b + (long)dz * HW;

      // Stage 24x22 input tile (h in [hbase-3, hbase+18], w in [wbase-3, wbase+18])
      for (int i = lane; i < 24 * 22; i += 32) {
        const int row = i / 22, col = i % 22;
        const int h = hbase + row - 3, w = wbase + col - 3;
        float v = 0.0f;
        if (row < 22 && h >= 0 && h < H_ && w >= 0 && w < W_)
          v = plane[h * W_ + w];
        tile[row * TW + col] = v;
      }
      __syncthreads();

      for (int kw = 0; kw < 7; ++kw) {
        const int wb0 = (c * 7 + kd) * 49 + kw;  // kh stride = 7
        // Per-lane nonzero Toeplitz entries: slot s has K-index ks=kgrp+s;
        // valid taps are kh = r and kh = r+4 (r = (ks-m) mod 4), appearing at
        // chunks c0 and c0+1 respectively.
        float w0[2], w1[2]; int c0[2];
#pragma unroll
        for (int s = 0; s < 2; ++s) {
          const int ks = kgrp + s;
          const int r  = (ks - m) & 3;
          c0[s] = (m + r - ks) >> 2;
          w0[s] = lw[wb0 + r * 7];
          w1[s] = (r <= 2) ? lw[wb0 + (r + 4) * 7] : 0.0f;
        }
#pragma unroll
        for (int ch = 0; ch < 6; ++ch) {
          v2f a, bb;
          a.x = (ch == c0[0]) ? w0[0] : ((ch == c0[0] + 1) ? w1[0] : 0.0f);
          a.y = (ch == c0[1]) ? w0[1] : ((ch == c0[1] + 1) ? w1[1] : 0.0f);
          const int j0 = ch * 4 + kgrp;          // tile rows for this lane's K pair
          bb.x = tile[j0 * TW + kw + n];
          bb.y = tile[(j0 + 1) * TW + kw + n];
          acc = __builtin_amdgcn_wmma_f32_16x16x4_f32(
              false, a, false, bb, (short)0, acc, false, false);
        }
      }
      __syncthreads();                            // before tile is overwritten
    }
  }

  const float bv = bias[0];
  float* sp = scale + ((long)b * D_ + d) * HW;
#pragma unroll
  for (int r = 0; r < 8; ++r) {
    const int M = r + (kgrp << 2);               // +0 or +8
    const float y = acc[r] + bv;
    sp[(hbase + M) * W_ + (wbase + n)] = 1.0f / (1.0f + __expf(-y));
  }
}

// ---------------------------------------------------------------------------
// Kernel 3: out = x * sigmoid(conv(...)), broadcast over C.  Pure bandwidth.
// ---------------------------------------------------------------------------
__global__ __launch_bounds__(256) void sg_apply(
    const float* __restrict__ x, const float* __restrict__ scale,
    float* __restrict__ out) {
  long t = (long)blockIdx.x * blockDim.x + threadIdx.x;   // 8388608 threads
  long f = t * 4;
  int  b = (int)(f / CDHW);
  int  p = (int)((f % CDHW) % DHW);
  float4 xv = *(const float4*)(x + f);
  float4 sv = *(const float4*)(scale + (long)b * DHW + p);
  float4 o;
  o.x = xv.x * sv.x; o.y = xv.y * sv.y;
  o.z = xv.z * sv.z; o.w = xv.w * sv.w;
  *(float4*)(out + f) = o;
}

// ---------------------------------------------------------------------------
extern "C" void kernel_launch(void* const* d_in, const int* in_sizes, int n_in,
                              void* d_out, int out_size, void* d_ws, size_t ws_size,
                              hipStream_t stream) {
  const float* x      = (const float*)d_in[0];   // [4,64,32,64,64]
  const float* conv_w = (const float*)d_in[1];   // [1,2,7,7,7] = 686
  const float* conv_b = (const float*)d_in[2];   // [1]
  float* out = (float*)d_out;

  float* ws    = (float*)d_ws;
  float* xmax  = ws;                 // B*DHW = 524288 floats (2 MiB)
  float* xmean = ws + (long)B_ * DHW;
  float* scale = ws + 2L * B_ * DHW; // another 2 MiB

  // Phase 1: channel reduce (524288 outputs, 4 per thread)
  sg_reduce_maxmean<<<512, 256, 0, stream>>>(x, xmax, xmean);

  // Phase 2: WMMA Toeplitz conv + sigmoid (2048 tiles, 1 wave each)
  sg_conv_sigmoid<<<2048, 32, 0, stream>>>(xmax, xmean, conv_w, conv_b, scale);

  // Phase 3: broadcast multiply (33554432 elements, 4 per thread)
  sg_apply<<<32768, 256, 0, stream>>>(x, scale, out);
}